// MyModel_46256797778100
// MI455X (gfx1250) — compile-verified
//
#include <hip/hip_runtime.h>

// Ball query (PointNet++) for MI455X / gfx1250, wave32.
//
// d2(m,n) = |q_m|^2 + |p_n|^2 - 2 q_m.p_n  computed as one
// V_WMMA_F32_16X16X4_F32 per 16x16 (query,point) tile:
//   A(16x4)  row m = (-2qx, -2qy, -2qz, 1)        [loaded once per wave]
//   B(4x16)  col n = ( px,   py,   pz,  |p|^2 )   [from LDS-staged strip]
//   C(16x16) row m = |q_m|^2 broadcast
// Point data is double-buffered in LDS via GLOBAL_LOAD_ASYNC_TO_LDS_B128
// (ASYNCcnt-tracked), overlapping the next strip's fetch with the current
// strip's WMMA work. 8 wave-uniform ballots per tile turn the 256 compares
// into 16-bit hit masks; owner lanes 0..15 do the sequential slot recording.

typedef __attribute__((ext_vector_type(2))) float v2f;
typedef __attribute__((ext_vector_type(8))) float v8f;

#define RADIUS2   0.01f
#define NSAMPLE   64
#define TILE_M    16
#define TILE_N    16
#define STRIP     128                  // points per async-staged strip
#define OPS_PER_STRIP 3                // 3 x b128 x 32 lanes = 1536 B = STRIP*12

struct SMem {
    float pts[2][STRIP * 3];           // 2 x 1536 B point strips
    int   idx[TILE_M * NSAMPLE];       // 4 KB output staging
};

// Raw 32-bit LDS byte offset of a __shared__ object (addrspace(3) ptrtoint).
__device__ __forceinline__ unsigned lds_off(const void* p) {
    return (unsigned)(unsigned long long)
        (__attribute__((address_space(3))) const void*)p;
}

// Issue one strip: 3 async b128 transfers (each lane moves 16 B per op).
__device__ __forceinline__ void async_strip(const float* gbase, unsigned ldsbase,
                                            int lane) {
    unsigned long long g = (unsigned long long)gbase;
#pragma unroll
    for (int j = 0; j < OPS_PER_STRIP; ++j) {
        unsigned off = (unsigned)(lane * 16 + j * 512);
        unsigned dst = ldsbase + off;
        asm volatile("global_load_async_to_lds_b128 %0, %1, %2"
                     :: "v"(dst), "v"(off), "s"(g) : "memory");
    }
}

__global__ __launch_bounds__(32) void ball_query_wmma_kernel(
    const float* __restrict__ new_xyz,   // [B*M*3]
    const float* __restrict__ xyz,       // [B*N*3]
    int* __restrict__ out,               // [B*M*NSAMPLE]
    int M, int N)
{
    __shared__ SMem sm;

    const int lane    = threadIdx.x;            // 0..31 (wave32)
    const int gq0     = blockIdx.x * TILE_M;    // flat query-tile base in [0, B*M)
    const int b       = gq0 / M;                // batch (tiles never cross batches)
    const int m_local = lane & 15;
    const int gq      = gq0 + m_local;          // this lane's query (flat index)

    // ---- load this lane's query, build A and C (once per wave) ----
    const float qx = new_xyz[3 * gq + 0];
    const float qy = new_xyz[3 * gq + 1];
    const float qz = new_xyz[3 * gq + 2];
    const float qq = qx * qx + qy * qy + qz * qz;

    // A layout (16x4 f32): lanes 0-15 hold K=0,1; lanes 16-31 hold K=2,3.
    v2f A;
    if (lane < 16) { A[0] = -2.0f * qx; A[1] = -2.0f * qy; }
    else           { A[0] = -2.0f * qz; A[1] = 1.0f;       }

    // C layout (16x16 f32): VGPR r = row M=r (lanes 0-15) / M=r+8 (lanes 16-31).
    v8f C;
#pragma unroll
    for (int r = 0; r < 8; ++r)
        C[r] = __shfl(qq, (lane < 16) ? r : (r + 8), 32);

    const float* __restrict__ pbase = xyz + (size_t)b * N * 3;
    const unsigned pbuf_off[2] = { lds_off(&sm.pts[0][0]), lds_off(&sm.pts[1][0]) };

    int  cnt   = (lane < 16) ? 0 : NSAMPLE;   // non-owner lanes start "done"
    int  first = 0;
    bool done  = false;

    const int nstrips = N / STRIP;

    // prime the pipeline: strip 0 -> buffer 0
    async_strip(pbase, pbuf_off[0], lane);

    for (int s = 0; s < nstrips && !done; ++s) {
        const bool have_next = (s + 1 < nstrips);
        if (have_next)
            async_strip(pbase + (size_t)(s + 1) * STRIP * 3,
                        pbuf_off[(s + 1) & 1], lane);

        // In-order async completion: after the wait, strip s is resident.
        if (have_next) asm volatile("s_wait_asynccnt 0x3" ::: "memory");
        else           asm volatile("s_wait_asynccnt 0x0" ::: "memory");

        const float* sp = sm.pts[s & 1];

#pragma unroll 2
        for (int c = 0; c < STRIP / TILE_N; ++c) {
            // ---- 16 points from LDS (lane reads point c*16 + (lane&15)) ----
            const int pidx = (c * TILE_N + m_local) * 3;
            const float px = sp[pidx + 0];
            const float py = sp[pidx + 1];
            const float pz = sp[pidx + 2];
            const float pp = px * px + py * py + pz * pz;

            // B layout (4x16 f32): lanes 0-15 hold K=0,1; lanes 16-31 hold K=2,3.
            v2f Bm;
            if (lane < 16) { Bm[0] = px; Bm[1] = py; }
            else           { Bm[0] = pz; Bm[1] = pp; }

            // D = A x B + C  ->  16x16 tile of squared distances
            v8f D = __builtin_amdgcn_wmma_f32_16x16x4_f32(
                /*neg_a=*/false, A, /*neg_b=*/false, Bm,
                /*c_mod=*/(short)0, C, /*reuse_a=*/false, /*reuse_b=*/false);

            // ---- 8 wave-uniform ballots: hit masks for all 256 pairs ----
            unsigned bal[8];
#pragma unroll
            for (int r = 0; r < 8; ++r)
                bal[r] = __builtin_amdgcn_ballot_w32(D[r] < RADIUS2);

            // Owner lane L (<16) wants bal[L&7]; queries 0-7 in low 16 bits,
            // queries 8-15 in high 16 bits.
            unsigned bv = 0;
#pragma unroll
            for (int r = 0; r < 8; ++r)
                bv = ((lane & 7) == r) ? bal[r] : bv;
            unsigned bits = (lane < 8)  ? (bv & 0xFFFFu)
                          : (lane < 16) ? (bv >> 16)
                                        : 0u;

            // ---- sequential slot recording (reference semantics) ----
            const int base_n = s * STRIP + c * TILE_N;
            while (bits) {
                const int bpos = __builtin_ctz(bits);
                bits &= bits - 1;
                const int idx = base_n + bpos;
                if (cnt == 0) first = idx;
                if (cnt < NSAMPLE) sm.idx[m_local * NSAMPLE + cnt] = idx;
                ++cnt;
            }

            // wave-uniform early exit once every owned query has 64 hits
            if (__builtin_amdgcn_ballot_w32(cnt < NSAMPLE) == 0u) {
                done = true;
                break;
            }
        }
    }

    // ---- fill remaining slots with the first hit index (0 if none) ----
    if (lane < 16) {
        for (int k = cnt; k < NSAMPLE; ++k)
            sm.idx[m_local * NSAMPLE + k] = first;
    }

    __syncthreads();

    // ---- coalesced b128 store of the 16x64 index block ----
    const int4* __restrict__ lsrc = (const int4*)sm.idx;
    int4* __restrict__ dst = (int4*)(out + (size_t)gq0 * NSAMPLE);
#pragma unroll
    for (int i = 0; i < (TILE_M * NSAMPLE / 4) / 32; ++i)
        dst[i * 32 + lane] = lsrc[i * 32 + lane];
}

extern "C" void kernel_launch(void* const* d_in, const int* in_sizes, int n_in,
                              void* d_out, int out_size, void* d_ws, size_t ws_size,
                              hipStream_t stream) {
    const float* new_xyz = (const float*)d_in[0];   // [B, M, 3] f32
    const float* xyz     = (const float*)d_in[1];   // [B, N, 3] f32
    int* out = (int*)d_out;                         // [B, M, 64] i32

    const int B = 2;
    const int M = in_sizes[0] / (B * 3);            // 4096
    const int N = in_sizes[1] / (B * 3);            // 16384

    const int ntiles = (B * M) / TILE_M;            // 512 waves, one per 16 queries
    ball_query_wmma_kernel<<<ntiles, 32, 0, stream>>>(new_xyz, xyz, out, M, N);
}